// Pytorch_filter_smoother_Obj_42588895707544
// MI455X (gfx1250) — compile-verified
//
#include <hip/hip_runtime.h>
#include <cstddef>

#define NSTEP 4096
#define BTOT  256
#define DX_   32
#define DZ_   16
#define LSX   36   // padded LDS row stride (floats) for 32-row tiles -> conflict-free b128/b64
#define LSH   20   // padded LDS row stride for 16-row tiles

typedef float v2f __attribute__((ext_vector_type(2)));
typedef float v4f __attribute__((ext_vector_type(4)));
typedef float v8f __attribute__((ext_vector_type(8)));

// D = A(16x4 f32) * B(4x16 f32) + C ; full-rate f32 tensor op on CDNA5
__device__ __forceinline__ v8f wmma_f32(v2f a, v2f b, v8f c) {
  return __builtin_amdgcn_wmma_f32_16x16x4_f32(false, a, false, b, (short)0, c,
                                               false, false);
}
__device__ __forceinline__ v2f ld2(const float* p) { return *(const v2f*)p; }
__device__ __forceinline__ void st2(float* p, v2f v) { *(v2f*)p = v; }

// ---------------------------------------------------------------------------
// Forward filter: xs[n] = x_n ;  x_{n+1} = F x_n + K z_n
// One workgroup = 16 batch columns, 2 waves (wave t owns output rows 16t..16t+15).
// State X kept in WMMA B-operand layout; D->B relayout via LDS each step.
// Time loop unrolled by 2 so all ping/pong indices (n&1) are compile-time.
// ---------------------------------------------------------------------------
__global__ __launch_bounds__(64) void fwd_filter(
    const float* __restrict__ z, const float* __restrict__ x0,
    const float* __restrict__ F, const float* __restrict__ Km,
    float* __restrict__ xs) {
  __shared__ float lbuf[2][16 * LSX];

  const int lane = threadIdx.x & 31;
  const int wv   = threadIdx.x >> 5;  // row-tile index (wave-uniform)
  const int col  = lane & 15;         // batch column within group / M row in tile
  const int h    = lane >> 4;
  const int b0   = blockIdx.x * 16;

  // A-operand tiles of F (8 K-chunks) and K (4 K-chunks):
  // lane holds M = 16*wv+col, K = 4c+2h, 4c+2h+1 (contiguous pair -> b64)
  v2f FA[8], KA[4], X[8];
#pragma unroll
  for (int c = 0; c < 8; ++c)
    FA[c] = ld2(&F[(wv * 16 + col) * DX_ + 4 * c + 2 * h]);
#pragma unroll
  for (int c = 0; c < 4; ++c)
    KA[c] = ld2(&Km[(wv * 16 + col) * DZ_ + 4 * c + 2 * h]);
  // initial state in B layout: chunk c -> rows 4c+2h,+1 at column col
#pragma unroll
  for (int c = 0; c < 8; ++c)
    X[c] = ld2(&x0[(size_t)(b0 + col) * DX_ + 4 * c + 2 * h]);

  v2f zb[2][4];
#pragma unroll
  for (int c = 0; c < 4; ++c)
    zb[0][c] = ld2(&z[(size_t)(b0 + col) * DZ_ + 4 * c + 2 * h]);

#pragma unroll 2
  for (int n = 0; n < NSTEP; ++n) {
    const int cur = n & 1;            // compile-time after unroll-by-2
    const int nxt = cur ^ 1;
    if (n + 1 < NSTEP) {  // prefetch z[n+1]
      const float* zp = &z[((size_t)(n + 1) * BTOT + b0 + col) * DZ_];
#pragma unroll
      for (int c = 0; c < 4; ++c) zb[nxt][c] = ld2(&zp[4 * c + 2 * h]);
    }
    {  // emit xs[n] = pre-update state; each wave stores 4 of the 8 chunks.
      // Branch on the wave-uniform wv so all register-array indices are
      // compile-time constants (no cndmask select trees).
      float* o = &xs[((size_t)n * BTOT + b0 + col) * DX_];
      if (wv == 0) {
#pragma unroll
        for (int c = 0; c < 4; ++c) st2(&o[4 * c + 2 * h], X[c]);
      } else {
#pragma unroll
        for (int c = 0; c < 4; ++c) st2(&o[16 + 4 * c + 2 * h], X[c + 4]);
      }
    }
    v8f acc = {};
#pragma unroll
    for (int c = 0; c < 4; ++c) acc = wmma_f32(KA[c], zb[cur][c], acc);  // K z
#pragma unroll
    for (int c = 0; c < 8; ++c) acc = wmma_f32(FA[c], X[c], acc);        // + F x

    // relayout D (C/D layout) -> B layout via double-buffered LDS
    float* buf = lbuf[cur];
    const int base = col * LSX + wv * 16 + 8 * h;
    *(v4f*)&buf[base]     = __builtin_shufflevector(acc, acc, 0, 1, 2, 3);
    *(v4f*)&buf[base + 4] = __builtin_shufflevector(acc, acc, 4, 5, 6, 7);
    __syncthreads();
#pragma unroll
    for (int c = 0; c < 8; ++c)
      X[c] = ld2(&buf[col * LSX + 4 * c + 2 * h]);
  }
}

// ---------------------------------------------------------------------------
// Backward smoother: bz = z - H^T xs ; s = F^T s + Sint[:, :DZ] bz ;
//                    out2 = xs + Sigma s
// ---------------------------------------------------------------------------
__global__ __launch_bounds__(64) void bwd_smoother(
    const float* __restrict__ z, const float* __restrict__ F,
    const float* __restrict__ H, const float* __restrict__ Sint,
    const float* __restrict__ Sigma, const float* __restrict__ xs,
    float* __restrict__ out2) {
  __shared__ float lbufS[2][16 * LSX];
  __shared__ float lbufH[2][16 * LSH];  // per-wave 16x16 scratch

  const int lane = threadIdx.x & 31;
  const int wv   = threadIdx.x >> 5;
  const int col  = lane & 15;
  const int h    = lane >> 4;
  const int b0   = blockIdx.x * 16;

  v2f FTA[8], HTA[8], SIA[4], SGA[8], S[8];
#pragma unroll
  for (int c = 0; c < 8; ++c) {
    v2f t;
    t.x = F[(4 * c + 2 * h) * DX_ + wv * 16 + col];      // F^T A-tile
    t.y = F[(4 * c + 2 * h + 1) * DX_ + wv * 16 + col];
    FTA[c] = t;
    t.x = H[(4 * c + 2 * h) * DZ_ + col];                // H^T A-tile (16 rows)
    t.y = H[(4 * c + 2 * h + 1) * DZ_ + col];
    HTA[c] = t;
    SGA[c] = ld2(&Sigma[(wv * 16 + col) * DX_ + 4 * c + 2 * h]);
    t.x = 0.f; t.y = 0.f;
    S[c] = t;  // s[N] = 0
  }
#pragma unroll
  for (int c = 0; c < 4; ++c)  // Sint[:, :16] A-tiles
    SIA[c] = ld2(&Sint[(wv * 16 + col) * DX_ + 4 * c + 2 * h]);

  v2f zb[2][4], xb[2][8];
  {
    const float* zp = &z[((size_t)(NSTEP - 1) * BTOT + b0 + col) * DZ_];
    const float* xp = &xs[((size_t)(NSTEP - 1) * BTOT + b0 + col) * DX_];
#pragma unroll
    for (int c = 0; c < 4; ++c) zb[(NSTEP - 1) & 1][c] = ld2(&zp[4 * c + 2 * h]);
#pragma unroll
    for (int c = 0; c < 8; ++c) xb[(NSTEP - 1) & 1][c] = ld2(&xp[4 * c + 2 * h]);
  }

#pragma unroll 2
  for (int n = NSTEP - 1; n >= 0; --n) {
    const int cur = n & 1;            // compile-time after unroll-by-2
    const int nxt = cur ^ 1;
    if (n > 0) {  // prefetch z[n-1], xs[n-1] (xs is L2-resident: 128MB < 192MB)
      const float* zp = &z[((size_t)(n - 1) * BTOT + b0 + col) * DZ_];
      const float* xp = &xs[((size_t)(n - 1) * BTOT + b0 + col) * DX_];
#pragma unroll
      for (int c = 0; c < 4; ++c) zb[nxt][c] = ld2(&zp[4 * c + 2 * h]);
#pragma unroll
      for (int c = 0; c < 8; ++c) xb[nxt][c] = ld2(&xp[4 * c + 2 * h]);
    }

    // hx = H^T xs[n]  (16x16; computed by both waves, per-wave LDS relayout)
    v8f hx = {};
#pragma unroll
    for (int c = 0; c < 8; ++c) hx = wmma_f32(HTA[c], xb[cur][c], hx);
    float* hb = lbufH[wv];
    const int hbase = col * LSH + 8 * h;
    *(v4f*)&hb[hbase]     = __builtin_shufflevector(hx, hx, 0, 1, 2, 3);
    *(v4f*)&hb[hbase + 4] = __builtin_shufflevector(hx, hx, 4, 5, 6, 7);
    v2f bz[4];
#pragma unroll
    for (int c = 0; c < 4; ++c)
      bz[c] = zb[cur][c] - ld2(&hb[col * LSH + 4 * c + 2 * h]);

    // s_new = Sint[:, :16] bz + F^T s
    v8f sa = {};
#pragma unroll
    for (int c = 0; c < 4; ++c) sa = wmma_f32(SIA[c], bz[c], sa);
#pragma unroll
    for (int c = 0; c < 8; ++c) sa = wmma_f32(FTA[c], S[c], sa);

    float* sbuf = lbufS[cur];
    const int base = col * LSX + wv * 16 + 8 * h;
    *(v4f*)&sbuf[base]     = __builtin_shufflevector(sa, sa, 0, 1, 2, 3);
    *(v4f*)&sbuf[base + 4] = __builtin_shufflevector(sa, sa, 4, 5, 6, 7);
    __syncthreads();
#pragma unroll
    for (int c = 0; c < 8; ++c)
      S[c] = ld2(&sbuf[col * LSX + 4 * c + 2 * h]);

    // out2[n] = xs[n] + Sigma s  (accumulate in C/D layout, add xs re-read
    // in matching layout -> contiguous b128 traffic, nontemporal store)
    v8f acc2 = {};
#pragma unroll
    for (int c = 0; c < 8; ++c) acc2 = wmma_f32(SGA[c], S[c], acc2);

    const float* xr = &xs[((size_t)n * BTOT + b0 + col) * DX_ + wv * 16 + 8 * h];
    v4f xlo = *(const v4f*)xr;
    v4f xhi = *(const v4f*)(xr + 4);
    v4f olo = __builtin_shufflevector(acc2, acc2, 0, 1, 2, 3) + xlo;
    v4f ohi = __builtin_shufflevector(acc2, acc2, 4, 5, 6, 7) + xhi;
    float* op = &out2[((size_t)n * BTOT + b0 + col) * DX_ + wv * 16 + 8 * h];
    __builtin_nontemporal_store(olo, (v4f*)op);
    __builtin_nontemporal_store(ohi, (v4f*)(op + 4));
  }
}

// ---------------------------------------------------------------------------
extern "C" void kernel_launch(void* const* d_in, const int* in_sizes, int n_in,
                              void* d_out, int out_size, void* d_ws,
                              size_t ws_size, hipStream_t stream) {
  (void)in_sizes; (void)n_in; (void)out_size; (void)d_ws; (void)ws_size;
  const float* z     = (const float*)d_in[0];
  const float* x0    = (const float*)d_in[1];
  const float* F     = (const float*)d_in[2];
  const float* Km    = (const float*)d_in[3];
  const float* H     = (const float*)d_in[4];
  const float* Sint  = (const float*)d_in[5];
  const float* Sigma = (const float*)d_in[6];

  float* xs  = (float*)d_out;                       // output 0: (N,B,DX,1)
  float* hat = xs + (size_t)NSTEP * BTOT * DX_;     // output 1: (N,B,DX,1)

  fwd_filter<<<dim3(BTOT / 16), dim3(64), 0, stream>>>(z, x0, F, Km, xs);
  bwd_smoother<<<dim3(BTOT / 16), dim3(64), 0, stream>>>(z, F, H, Sint, Sigma,
                                                         xs, hat);
}